// sparse3DBA_89481348645122
// MI455X (gfx1250) — compile-verified
//
#include <hip/hip_runtime.h>
#include <math.h>

// ---------------- workspace layout (floats) ----------------
#define WS_R    0     // current R (9, row-major)
#define WS_T    9     // current t (3)
#define WS_LAM  12
#define WS_LR   13
#define WS_PREV 14
#define WS_RN   16    // proposed R (9)
#define WS_TN   25    // proposed t (3)
#define WS_ACC  32    // 7x7 augmented normal-equation accumulator (49)
#define WS_COST 96    // masked cost numerator
#define WS_CNT  97    // mask count

typedef __attribute__((ext_vector_type(2))) float v2f;
typedef __attribute__((ext_vector_type(8))) float v8f;

__device__ __forceinline__ int floordiv_i(long long a, long long b) {
  long long q = a / b;
  if ((a % b) != 0 && ((a < 0) != (b < 0))) --q;
  return (int)q;
}

__device__ __forceinline__ void project_pt(
    const float* __restrict__ R, const float* __restrict__ t,
    float fx, float fy, float cx, float cy,
    float px, float py, float pz,
    float& X, float& Y, float& Z, int& ix, int& iy)
{
  X = R[0]*px + R[1]*py + R[2]*pz + t[0];
  Y = R[3]*px + R[4]*py + R[5]*pz + t[1];
  Z = R[6]*px + R[7]*py + R[8]*pz + t[2];
  float u  = fx*X + cx*Z;
  float v  = fy*Y + cy*Z;
  float sx = fminf(fmaxf(u / Z, -1.0e6f), 1.0e6f);
  float sy = fminf(fmaxf(v / Z, -1.0e6f), 1.0e6f);
  ix = (int)rintf(sx) - 1;
  iy = (int)rintf(sy) - 1;
}

__device__ __forceinline__ void gather_rc(int ix, int iy, int W, int H,
                                          int Wf, int Hf, int& row, int& col)
{
  int r = floordiv_i((long long)iy * (long long)Hf, (long long)H);
  int c = floordiv_i((long long)ix * (long long)Wf, (long long)W);
  row = min(max(r, 0), Hf - 1);
  col = min(max(c, 0), Wf - 1);
}

// -----------------------------------------------------------------------
// Derivative pass: per wave, one point at a time. 32 lanes each own one
// channel of a 32-channel chunk, build augmented row r = m*[J(6), err],
// stage to LDS, and feed V_WMMA_F32_16X16X4_F32 to accumulate sum(r^T r)
// (Hess in [0..5]x[0..5], Grad in column 6) in a resident 16x16 f32 tile.
// -----------------------------------------------------------------------
__global__ void __launch_bounds__(256)
accum_kernel(const float* __restrict__ pts, const float* __restrict__ ref,
             const float* __restrict__ fm,  const float* __restrict__ gxm,
             const float* __restrict__ gym, const float* __restrict__ Km,
             const int* __restrict__ pW, const int* __restrict__ pH,
             float* __restrict__ ws, int N, int C)
{
  __shared__ float stage[8][32][16];

  const int lane = threadIdx.x & 31;
  const int wv   = threadIdx.x >> 5;
  const int wavesPerBlock = blockDim.x >> 5;
  const int gwave  = blockIdx.x * wavesPerBlock + wv;
  const int nwaves = gridDim.x * wavesPerBlock;

  const int W = *pW, H = *pH;
  const int Hf = H, Wf = W;           // feature map matches image extent here
  const int HW = Hf * Wf;
  const float fx = Km[0], cx = Km[2], fy = Km[4], cy = Km[5];

  float Rl[9], tl[3];
#pragma unroll
  for (int i = 0; i < 9; ++i) Rl[i] = ws[WS_R + i];
#pragma unroll
  for (int i = 0; i < 3; ++i) tl[i] = ws[WS_T + i];

  // zero the padding columns once (rows are overwritten every chunk)
#pragma unroll
  for (int j = 7; j < 16; ++j) stage[wv][lane][j] = 0.0f;

  const int half = lane >> 4;
  const int m16  = lane & 15;
  const int nchunks = C >> 5;          // C assumed multiple of 32 (C=128)

  v8f acc = {};

  for (int n = gwave; n < N; n += nwaves) {
    const float px = pts[3*n+0], py = pts[3*n+1], pz = pts[3*n+2];
    float X, Y, Z; int ix, iy;
    project_pt(Rl, tl, fx, fy, cx, cy, px, py, pz, X, Y, Z, ix, iy);

    const float m = (ix >= 0 && iy >= 0 && ix < W && iy < H) ? 1.0f : 0.0f;
    int row, col;
    gather_rc(ix, iy, W, H, Wf, Hf, row, col);
    const int base = row * Wf + col;

    const float Zs = (fabsf(Z) > 1e-6f) ? Z : 1e-6f;
    const float u  = fx / Zs,  v  = fy / Zs;
    const float xz = X / Zs,   yz = Y / Zs;
    // A = J_px_p @ [I | -skew(p3)]  (2 x 6), channel-independent
    const float A0[6] = { u, 0.0f, -u*xz, -u*xz*Y,      u*Z + u*xz*X, -u*Y };
    const float A1[6] = { 0.0f, v, -v*yz, -v*(Z+yz*Y),  v*yz*X,        v*X };

    for (int cc = 0; cc < nchunks; ++cc) {
      const int c = (cc << 5) + lane;
      const float f  = fm [c*HW + base];
      const float gx = gxm[c*HW + base];
      const float gy = gym[c*HW + base];
      const float e  = f - ref[(size_t)n * C + c];

#pragma unroll
      for (int j = 0; j < 6; ++j)
        stage[wv][lane][j] = m * (gx * A0[j] + gy * A1[j]);
      stage[wv][lane][6] = m * e;

      // LDS stores must land before the cross-lane fragment loads (DScnt).
      asm volatile("s_wait_dscnt 0x0" ::: "memory");

#pragma unroll
      for (int kb = 0; kb < 32; kb += 4) {
        // 16x4 f32 A fragment: lanes 0-15 hold K=kb,kb+1; lanes 16-31 K=kb+2,kb+3
        v2f a;
        a.x = stage[wv][kb + 2*half + 0][m16];
        a.y = stage[wv][kb + 2*half + 1][m16];
        // symmetric SYRK: B fragment layout mirrors A for r^T r, so b == a
        acc = __builtin_amdgcn_wmma_f32_16x16x4_f32(
            /*neg_a=*/false, a, /*neg_b=*/false, a,
            /*c_mod=*/(short)0, acc, /*reuse_a=*/false, /*reuse_b=*/false);
      }
    }
  }

  // Flush the 7x7 corner of the 16x16 accumulator (M = v + 8*half, N = m16)
#pragma unroll
  for (int vI = 0; vI < 8; ++vI) {
    const int Mrow = vI + 8 * half;
    if (Mrow < 7 && m16 < 7)
      atomicAdd(&ws[WS_ACC + Mrow * 7 + m16], acc[vI]);
  }
}

// ---------------- masked cost at a given state (offset selects R,t) -----
__global__ void __launch_bounds__(256)
cost_kernel(const float* __restrict__ pts, const float* __restrict__ ref,
            const float* __restrict__ fm,  const float* __restrict__ Km,
            const int* __restrict__ pW, const int* __restrict__ pH,
            float* __restrict__ ws, int stateOff, int N, int C)
{
  const int n = blockIdx.x * blockDim.x + threadIdx.x;
  const int W = *pW, H = *pH;
  const int Hf = H, Wf = W;
  const int HW = Hf * Wf;
  const float fx = Km[0], cx = Km[2], fy = Km[4], cy = Km[5];

  float part = 0.0f, cnt = 0.0f;
  if (n < N) {
    float Rl[9], tl[3];
#pragma unroll
    for (int i = 0; i < 9; ++i) Rl[i] = ws[stateOff + i];
#pragma unroll
    for (int i = 0; i < 3; ++i) tl[i] = ws[stateOff + 9 + i];

    const float px = pts[3*n+0], py = pts[3*n+1], pz = pts[3*n+2];
    float X, Y, Z; int ix, iy;
    project_pt(Rl, tl, fx, fy, cx, cy, px, py, pz, X, Y, Z, ix, iy);
    const float m = (ix >= 0 && iy >= 0 && ix < W && iy < H) ? 1.0f : 0.0f;
    int row, col;
    gather_rc(ix, iy, W, H, Wf, Hf, row, col);
    const int base = row * Wf + col;

    float s = 0.0f;
    for (int c = 0; c < C; ++c) {
      const float e = fm[c*HW + base] - ref[(size_t)n * C + c];
      s += e * e;
    }
    part = 0.5f * s * m;
    cnt  = m;
  }
  // wave32 reduction
#pragma unroll
  for (int off = 16; off > 0; off >>= 1) {
    part += __shfl_down(part, off, 32);
    cnt  += __shfl_down(cnt,  off, 32);
  }
  if ((threadIdx.x & 31) == 0) {
    atomicAdd(&ws[WS_COST], part);
    atomicAdd(&ws[WS_CNT],  cnt);
  }
}

// ---------------- state init / tiny sequential kernels ------------------
__global__ void init_kernel(const float* __restrict__ R0,
                            const float* __restrict__ t0, float* __restrict__ ws)
{
  const int i = threadIdx.x;
  if (i < 9)  ws[WS_R + i] = R0[i];
  if (i < 3)  ws[WS_T + i] = t0[i];
  if (i == 0) { ws[WS_LAM] = 0.01f; ws[WS_LR] = 1.0f; ws[WS_PREV] = 0.0f; }
  if (i < 49) ws[WS_ACC + i] = 0.0f;
  if (i < 2)  ws[WS_COST + i] = 0.0f;
}

__global__ void init_cost_kernel(float* __restrict__ ws)
{
  ws[WS_PREV] = ws[WS_COST] / fmaxf(ws[WS_CNT], 1.0f);
  ws[WS_COST] = 0.0f; ws[WS_CNT] = 0.0f;
}

__device__ __forceinline__ void mat3mul(float* o, const float* a, const float* b)
{
#pragma unroll
  for (int i = 0; i < 3; ++i)
#pragma unroll
    for (int j = 0; j < 3; ++j)
      o[i*3+j] = a[i*3+0]*b[0*3+j] + a[i*3+1]*b[1*3+j] + a[i*3+2]*b[2*3+j];
}

__global__ void solve_kernel(float* __restrict__ ws)
{
  const float lam = ws[WS_LAM];
  const float lr  = ws[WS_LR];

  // damped normal equations from the WMMA accumulator
  float M[6][7];
#pragma unroll
  for (int i = 0; i < 6; ++i) {
#pragma unroll
    for (int j = 0; j < 6; ++j) M[i][j] = ws[WS_ACC + i*7 + j];
    M[i][6] = ws[WS_ACC + i*7 + 6];            // Grad
    M[i][i] += lam * (M[i][i] + 1e-9f);        // H + lam*diag(diag(H)+1e-9)
  }
  // Gauss-Jordan with partial pivoting: solve Hd x = Grad
  for (int k = 0; k < 6; ++k) {
    int piv = k; float best = fabsf(M[k][k]);
    for (int r = k + 1; r < 6; ++r)
      if (fabsf(M[r][k]) > best) { best = fabsf(M[r][k]); piv = r; }
    if (piv != k)
      for (int j = 0; j < 7; ++j) { float tmp = M[k][j]; M[k][j] = M[piv][j]; M[piv][j] = tmp; }
    float d = M[k][k];
    if (fabsf(d) < 1e-20f) d = (d < 0.0f) ? -1e-20f : 1e-20f;
    const float inv = 1.0f / d;
    for (int j = k; j < 7; ++j) M[k][j] *= inv;
    for (int r = 0; r < 6; ++r) {
      if (r == k) continue;
      const float f = M[r][k];
      for (int j = k; j < 7; ++j) M[r][j] -= f * M[k][j];
    }
  }
  float delta[6];
#pragma unroll
  for (int i = 0; i < 6; ++i) delta[i] = -lr * M[i][6];
  const float dtv[3] = { delta[0], delta[1], delta[2] };
  const float wx = delta[3], wy = delta[4], wz = delta[5];

  // so3exp(dw)
  const float th2 = wx*wx + wy*wy + wz*wz;
  const float th  = sqrtf(th2 + 1e-24f);
  float a, b;
  if (th < 1e-4f) { a = 1.0f - th2 / 6.0f; b = 0.5f - th2 / 24.0f; }
  else            { a = sinf(th) / th;     b = (1.0f - cosf(th)) / (th2 + 1e-24f); }
  const float Wm[9] = { 0, -wz, wy,  wz, 0, -wx,  -wy, wx, 0 };
  float W2[9]; mat3mul(W2, Wm, Wm);
  float dr[9];
#pragma unroll
  for (int i = 0; i < 9; ++i) dr[i] = a * Wm[i] + b * W2[i];
  dr[0] += 1.0f; dr[4] += 1.0f; dr[8] += 1.0f;

  float Rc[9], tc[3];
#pragma unroll
  for (int i = 0; i < 9; ++i) Rc[i] = ws[WS_R + i];
#pragma unroll
  for (int i = 0; i < 3; ++i) tc[i] = ws[WS_T + i];
  float Rn[9]; mat3mul(Rn, dr, Rc);
#pragma unroll
  for (int i = 0; i < 9; ++i) ws[WS_RN + i] = Rn[i];
#pragma unroll
  for (int i = 0; i < 3; ++i)
    ws[WS_TN + i] = dr[i*3+0]*tc[0] + dr[i*3+1]*tc[1] + dr[i*3+2]*tc[2] + dtv[i];

  ws[WS_COST] = 0.0f; ws[WS_CNT] = 0.0f;   // ready for the proposal cost pass
}

__global__ void update_kernel(float* __restrict__ ws)
{
  if (threadIdx.x == 0) {
    const float nc = ws[WS_COST] / fmaxf(ws[WS_CNT], 1.0f);
    const int accept = (nc <= ws[WS_PREV]) ? 1 : 0;
    ws[WS_LAM] = fminf(fmaxf(ws[WS_LAM] * (accept ? 0.1f : 10.0f), 1e-6f), 1e4f);
    ws[WS_LR]  = accept ? 1.0f : fminf(fmaxf(0.1f * ws[WS_LR], 0.001f), 1.0f);
    if (accept) {
      for (int i = 0; i < 9; ++i) ws[WS_R + i] = ws[WS_RN + i];
      for (int i = 0; i < 3; ++i) ws[WS_T + i] = ws[WS_TN + i];
      ws[WS_PREV] = nc;
    }
  }
  __syncthreads();
  const int i = threadIdx.x;
  if (i < 49) ws[WS_ACC + i] = 0.0f;
  if (i < 2)  ws[WS_COST + i] = 0.0f;
}

__global__ void writeout_kernel(const float* __restrict__ ws, float* __restrict__ out)
{
  const int i = threadIdx.x;
  if (i < 9) out[i]     = ws[WS_R + i];
  if (i < 3) out[9 + i] = ws[WS_T + i];
}

// -----------------------------------------------------------------------
extern "C" void kernel_launch(void* const* d_in, const int* in_sizes, int n_in,
                              void* d_out, int out_size, void* d_ws, size_t ws_size,
                              hipStream_t stream)
{
  const float* pts = (const float*)d_in[0];
  const float* ref = (const float*)d_in[1];
  const float* fm  = (const float*)d_in[2];
  const float* gxm = (const float*)d_in[3];
  const float* gym = (const float*)d_in[4];
  const float* Km  = (const float*)d_in[5];
  const float* R0  = (const float*)d_in[6];
  const float* t0  = (const float*)d_in[7];
  const int*   pW  = (const int*)d_in[8];
  const int*   pH  = (const int*)d_in[9];
  float* ws  = (float*)d_ws;
  float* out = (float*)d_out;

  const int N = in_sizes[0] / 3;
  const int C = in_sizes[1] / N;

  const int costBlocks = (N + 255) / 256;

  init_kernel<<<1, 64, 0, stream>>>(R0, t0, ws);
  cost_kernel<<<costBlocks, 256, 0, stream>>>(pts, ref, fm, Km, pW, pH, ws, WS_R, N, C);
  init_cost_kernel<<<1, 1, 0, stream>>>(ws);

  for (int it = 0; it < 10; ++it) {
    accum_kernel<<<256, 256, 0, stream>>>(pts, ref, fm, gxm, gym, Km, pW, pH, ws, N, C);
    solve_kernel<<<1, 1, 0, stream>>>(ws);
    cost_kernel<<<costBlocks, 256, 0, stream>>>(pts, ref, fm, Km, pW, pH, ws, WS_RN, N, C);
    update_kernel<<<1, 64, 0, stream>>>(ws);
  }

  writeout_kernel<<<1, 32, 0, stream>>>(ws, out);
}